// constrained_ouroboros_66168266162517
// MI455X (gfx1250) — compile-verified
//
#include <hip/hip_runtime.h>
#include <cstdint>
#include <cstddef>

// ---- problem constants (from reference) ----
#define B_    16
#define LRAW  4104
#define L_    4096      // LRAW - 8
#define TILES (L_/16)   // 256

// padded per-token strides (branch-free WMMA epilogues)
#define US  16          // u: cols 0-3 real, 4-15 scratch
#define PS  16          // pz: 0-3 xi_pre, 4-7 zgate, 8-15 scratch
#define SS  48          // sbuf: 0=dt_raw, 1..16=Bm, 17..32=Cm, 33..47 scratch

typedef float v2f __attribute__((ext_vector_type(2)));
typedef float v8f __attribute__((ext_vector_type(8)));

struct LayerP {
  const float* norm_w;     // (4)
  const float* in_proj_w;  // (4,8)
  const float* conv_w;     // (4,4)
  const float* conv_b;     // (4)
  const float* x_proj_w;   // (4,33)
  const float* dt_proj_w;  // (1,4)
  const float* dt_proj_b;  // (4)
  const float* A_log;      // (4,16)
  const float* Dv;         // (4)
  const float* out_proj_w; // (4,4)
};
struct LayerP3 { LayerP p[3]; };

struct NetBufs {
  float* u[3];     // (B,L,US)
  float* pz[3];    // (B,L,PS)
  float* xi[3];    // (B,L,4)
  float* sbuf[3];  // (B,L,SS)
  float* y[3];     // (B,L,4)
};

__device__ __forceinline__ float silu_f(float x) { return x / (1.f + __expf(-x)); }
__device__ __forceinline__ float softplus_f(float x) {
  return fmaxf(x, 0.f) + log1pf(__expf(-fabsf(x)));
}

// D = A(16x4) * B(4x16) + C, f32, wave32. One wave per 16-token tile.
__device__ __forceinline__ v8f wmma4(v2f a, v2f b, v8f c) {
  return __builtin_amdgcn_wmma_f32_16x16x4_f32(false, a, false, b, (short)0, c, false, false);
}

// ---------------- K1: derivative stencil, build inputs, init residuals ----------------
__global__ __launch_bounds__(256) void k_prep(const float* __restrict__ x,
                                              float* __restrict__ z1, float* __restrict__ xdot,
                                              float* __restrict__ u0, float* __restrict__ u1,
                                              float* __restrict__ u2) {
  int l = blockIdx.x * blockDim.x + threadIdx.x;
  int b = blockIdx.y;
  if (l >= L_) return;
  const float* xb = x + (size_t)b * LRAW;
  float xc = xb[l + 4];
  float xd = 0.8f * (xb[l + 5] - xb[l + 3]) - 0.2f * (xb[l + 6] - xb[l + 2])
           + (4.0f / 105.0f) * (xb[l + 7] - xb[l + 1]) - (1.0f / 280.0f) * (xb[l + 8] - xb[l + 0]);
  z1[(size_t)b * L_ + l] = xc;
  xdot[(size_t)b * L_ + l] = xd;
  size_t o  = ((size_t)b * L_ + l) * US;
  size_t of = ((size_t)b * L_ + (L_ - 1 - l)) * US;   // flipped half of x_in
  u0[o + 0] = xc; u0[o + 1] = xd; u0[of + 2] = xc; u0[of + 3] = xd;
  u1[o + 0] = xc; u1[o + 1] = xd; u1[of + 2] = xc; u1[of + 3] = xd;
  u2[o + 0] = xc; u2[o + 1] = xd; u2[of + 2] = xc; u2[of + 3] = xd;
}

// ---------------- K2: rmsnorm + in_proj (WMMA f32 16x16x4), branch-free epilogue ------
__global__ __launch_bounds__(32) void k_normproj(LayerP3 P, NetBufs nb) {
  int tile = blockIdx.x, b = blockIdx.y, net = blockIdx.z;
  const LayerP& p = P.p[net];
  int lane = threadIdx.x, row = lane & 15, half = lane >> 4;
  int l = tile * 16 + row;

  const float* u = nb.u[net] + ((size_t)b * L_) * US;
  float4 uv = *(const float4*)(u + (size_t)l * US);
  float ms = (uv.x * uv.x + uv.y * uv.y + uv.z * uv.z + uv.w * uv.w) * 0.25f;
  float s = rsqrtf(ms + 1e-5f);
  float t0 = uv.x * s * p.norm_w[0], t1 = uv.y * s * p.norm_w[1];
  float t2 = uv.z * s * p.norm_w[2], t3 = uv.w * s * p.norm_w[3];

  // A: lanes 0-15 hold K=0,1 ; lanes 16-31 hold K=2,3 (row M = lane&15)
  v2f a; a.x = half ? t2 : t0; a.y = half ? t3 : t1;
  // B (4x16, N padded 8->16): VGPR0 rows K0/K2, VGPR1 rows K1/K3 across half-waves
  int k0 = half ? 2 : 0;
  int col = row;
  v2f bb;
  bb.x = (col < 8) ? p.in_proj_w[(size_t)k0 * 8 + col] : 0.f;
  bb.y = (col < 8) ? p.in_proj_w[(size_t)(k0 + 1) * 8 + col] : 0.f;
  v8f c = {0.f, 0.f, 0.f, 0.f, 0.f, 0.f, 0.f, 0.f};
  v8f d = wmma4(a, bb, c);

  // full-tile store: cols 0-3 xi_pre, 4-7 zgate, 8-15 scratch (zeros)
  float* pz = nb.pz[net] + ((size_t)b * L_) * PS;
#pragma unroll
  for (int r = 0; r < 8; ++r) {
    int tok = tile * 16 + r + half * 8;
    pz[(size_t)tok * PS + col] = d[r];
  }
}

// ---------------- K3: causal conv + silu + x_proj (3x WMMA), branch-free epilogue -----
__global__ __launch_bounds__(32) void k_convproj(LayerP3 P, NetBufs nb) {
  int tile = blockIdx.x, b = blockIdx.y, net = blockIdx.z;
  const LayerP& p = P.p[net];
  int lane = threadIdx.x, row = lane & 15, half = lane >> 4;
  int l = tile * 16 + row;

  const float* pz = nb.pz[net] + ((size_t)b * L_) * PS;
  float xi_[4];
#pragma unroll
  for (int ch = 0; ch < 4; ++ch) {
    float acc = p.conv_b[ch];
#pragma unroll
    for (int k = 0; k < 4; ++k) {
      int idx = l + k - 3;
      float xv = (idx >= 0) ? pz[(size_t)idx * PS + ch] : 0.f;
      acc = fmaf(p.conv_w[ch * 4 + k], xv, acc);
    }
    xi_[ch] = silu_f(acc);
  }

  float* xib = nb.xi[net]   + ((size_t)b * L_) * 4;
  float* sb  = nb.sbuf[net] + ((size_t)b * L_) * SS;
  if (half == 0)
    *(float4*)(xib + (size_t)l * 4) = make_float4(xi_[0], xi_[1], xi_[2], xi_[3]);

  v2f a; a.x = half ? xi_[2] : xi_[0]; a.y = half ? xi_[3] : xi_[1];
  int k0 = half ? 2 : 0;
#pragma unroll
  for (int tn = 0; tn < 3; ++tn) {
    int colg = tn * 16 + (lane & 15);
    v2f bb;
    bb.x = (colg < 33) ? p.x_proj_w[(size_t)k0 * 33 + colg] : 0.f;
    bb.y = (colg < 33) ? p.x_proj_w[(size_t)(k0 + 1) * 33 + colg] : 0.f;
    v8f c = {0.f, 0.f, 0.f, 0.f, 0.f, 0.f, 0.f, 0.f};
    v8f d = wmma4(a, bb, c);
#pragma unroll
    for (int r = 0; r < 8; ++r) {
      int tok = tile * 16 + r + half * 8;
      sb[(size_t)tok * SS + colg] = d[r];   // cols 0..47, 33..47 are dead scratch
    }
  }
}

// ---------------- K4: selective scan; WG per (batch, net); lane=(e,n) ----------------
__global__ __launch_bounds__(64) void k_scan(LayerP3 P, NetBufs nb) {
  int b = blockIdx.x, net = blockIdx.y;
  const LayerP& p = P.p[net];
  int j = threadIdx.x, e = j >> 4, n = j & 15;

  float A_en = -__expf(p.A_log[e * 16 + n]);
  float dtw = p.dt_proj_w[e];
  float dtb = p.dt_proj_b[e];

  const float* sb = nb.sbuf[net] + ((size_t)b * L_) * SS;
  const float* xib = nb.xi[net] + ((size_t)b * L_) * 4;
  float* y = nb.y[net] + ((size_t)b * L_) * 4;

  float h = 0.f;
#pragma unroll 2
  for (int l = 0; l < L_; ++l) {
    const float* q = sb + (size_t)l * SS;
    float dtv = q[0];
    float del = softplus_f(fmaf(dtv, dtw, dtb));
    float Bn = q[1 + n], Cn = q[17 + n];
    float xie = xib[(size_t)l * 4 + e];
    float av = __expf(del * A_en);
    h = fmaf(av, h, del * Bn * xie);       // only loop-carried dependency
    float pv = h * Cn;
    pv += __shfl_xor(pv, 1, 32);
    pv += __shfl_xor(pv, 2, 32);
    pv += __shfl_xor(pv, 4, 32);
    pv += __shfl_xor(pv, 8, 32);           // sum over n within 16-lane group
    if (n == 0) y[(size_t)l * 4 + e] = pv;
  }
}

// ---------------- K5: skip-D + silu gate + out_proj (WMMA) + residual, branch-free ----
__global__ __launch_bounds__(32) void k_outproj(LayerP3 P, NetBufs nb) {
  int tile = blockIdx.x, b = blockIdx.y, net = blockIdx.z;
  const LayerP& p = P.p[net];
  int lane = threadIdx.x, row = lane & 15, half = lane >> 4;
  int l = tile * 16 + row;

  float4 y4  = *(const float4*)(nb.y[net]  + ((size_t)b * L_) * 4 + (size_t)l * 4);
  float4 xi4 = *(const float4*)(nb.xi[net] + ((size_t)b * L_) * 4 + (size_t)l * 4);
  float4 zg4 = *(const float4*)(nb.pz[net] + ((size_t)b * L_) * PS + (size_t)l * PS + 4);

  float g0 = (y4.x + p.Dv[0] * xi4.x) * silu_f(zg4.x);
  float g1 = (y4.y + p.Dv[1] * xi4.y) * silu_f(zg4.y);
  float g2 = (y4.z + p.Dv[2] * xi4.z) * silu_f(zg4.z);
  float g3 = (y4.w + p.Dv[3] * xi4.w) * silu_f(zg4.w);

  v2f a; a.x = half ? g2 : g0; a.y = half ? g3 : g1;
  int k0 = half ? 2 : 0;
  int col = row;
  v2f bb;
  bb.x = (col < 4) ? p.out_proj_w[(size_t)k0 * 4 + col] : 0.f;
  bb.y = (col < 4) ? p.out_proj_w[(size_t)(k0 + 1) * 4 + col] : 0.f;
  v8f c = {0.f, 0.f, 0.f, 0.f, 0.f, 0.f, 0.f, 0.f};
  v8f d = wmma4(a, bb, c);

  // unconditional residual RMW across full padded tile (cols>=4 add 0 to scratch)
  float* u = nb.u[net] + ((size_t)b * L_) * US;
#pragma unroll
  for (int r = 0; r < 8; ++r) {
    int tok = tile * 16 + r + half * 8;
    size_t ix = (size_t)tok * US + col;
    u[ix] = u[ix] + d[r];
  }
}

// ---------------- K6: smoothing + heads + yhat + concat outputs ----------------
__global__ __launch_bounds__(256) void k_heads(const float* __restrict__ u0,
                                               const float* __restrict__ u1,
                                               const float* __restrict__ u2,
                                               const float* __restrict__ z1,
                                               const float* __restrict__ xdot,
                                               const float* __restrict__ wo, const float* __restrict__ bo,
                                               const float* __restrict__ wg, const float* __restrict__ bg,
                                               const float* __restrict__ wd, const float* __restrict__ bd,
                                               const float* __restrict__ bsc, const float* __restrict__ dtp,
                                               float* __restrict__ out,
                                               float* __restrict__ ho, float* __restrict__ hg,
                                               float* __restrict__ hd) {
  int l = blockIdx.x * blockDim.x + threadIdx.x;
  int b = blockIdx.y;
  if (l >= L_) return;
  float dtv = dtp[0];
  int w = (int)roundf(0.001f / dtv);
  if (w < 1) w = 1;
  float inv = 1.f / (float)w;

  const float* m0 = u0 + ((size_t)b * L_) * US;
  const float* m1 = u1 + ((size_t)b * L_) * US;
  const float* m2 = u2 + ((size_t)b * L_) * US;

  float oc[4] = {0, 0, 0, 0}, gc[4] = {0, 0, 0, 0}, dc[4] = {0, 0, 0, 0};
  for (int i = 0; i < w; ++i) {
    int li = l + i - w / 2;
    li = li < 0 ? 0 : (li > L_ - 1 ? L_ - 1 : li);
    float4 a0 = *(const float4*)(m0 + (size_t)li * US);
    float4 a1 = *(const float4*)(m1 + (size_t)li * US);
    float4 a2 = *(const float4*)(m2 + (size_t)li * US);
    oc[0] += fabsf(a0.x); oc[1] += fabsf(a0.y); oc[2] += fabsf(a0.z); oc[3] += fabsf(a0.w);
    gc[0] += a1.x;        gc[1] += a1.y;        gc[2] += a1.z;        gc[3] += a1.w;
    dc[0] += fabsf(a2.x); dc[1] += fabsf(a2.y); dc[2] += fabsf(a2.z); dc[3] += fabsf(a2.w);
  }
  float omega = bo[0], gamma = bg[0], dd = bd[0];
#pragma unroll
  for (int ch = 0; ch < 4; ++ch) {
    oc[ch] *= inv; gc[ch] *= inv; dc[ch] *= inv;
    omega = fmaf(oc[ch], wo[ch], omega);
    gamma = fmaf(gc[ch], wg[ch], gamma);
    dd    = fmaf(dc[ch], wd[ch], dd);
  }
  gamma *= 1e-3f;                 // / TAU
  dd = fmaxf(dd, 0.f);
  float bb = fmaxf(bsc[0], 0.f) * 1e-3f;

  float z1v = z1[(size_t)b * L_ + l];
  float z2v = xdot[(size_t)b * L_ + l] / dtv;
  float yh = -omega * omega * z1v + gamma * z2v - bb * z1v * z1v * z2v - dd;

  out[(size_t)b * L_ + l] = yh;
  float* oc_out = out + (size_t)B_ * L_;
  size_t ob = ((size_t)b * L_ + l) * 12;
#pragma unroll
  for (int ch = 0; ch < 4; ++ch) {
    oc_out[ob + ch]     = oc[ch];
    oc_out[ob + 4 + ch] = gc[ch];
    oc_out[ob + 8 + ch] = dc[ch];
  }
  ho[(size_t)b * L_ + l] = omega;
  hg[(size_t)b * L_ + l] = gamma;
  hd[(size_t)b * L_ + l] = dd;
}

// ---------------- K7/K8: smoothness penalty (deterministic tree reductions) ----------------
__global__ __launch_bounds__(256) void k_pen1(const float* __restrict__ ho,
                                              const float* __restrict__ hg,
                                              const float* __restrict__ hd,
                                              float* __restrict__ pen) {
  int b = blockIdx.x, q = blockIdx.y;
  const float* u = (q == 0 ? ho : (q == 1 ? hg : hd)) + (size_t)b * L_;
  float s1 = 0.f, s2 = 0.f;
  for (int i = 1 + (int)threadIdx.x; i <= L_ - 2; i += (int)blockDim.x) {
    float dv = u[i + 1] - u[i];          // diff, sliced [1:]
    s1 += dv;
    s2 += dv * dv;
  }
  __shared__ float a1[256], a2[256];
  a1[threadIdx.x] = s1; a2[threadIdx.x] = s2;
  __syncthreads();
  for (int st = 128; st > 0; st >>= 1) {
    if ((int)threadIdx.x < st) {
      a1[threadIdx.x] += a1[threadIdx.x + st];
      a2[threadIdx.x] += a2[threadIdx.x + st];
    }
    __syncthreads();
  }
  if (threadIdx.x == 0) {
    float n = (float)(L_ - 2);           // 4094 samples
    float var = (a2[0] - a1[0] * a1[0] / n) / (n - 1.f);
    pen[q * B_ + b] = var;
  }
}

__global__ __launch_bounds__(64) void k_pen2(const float* __restrict__ pen,
                                             float* __restrict__ out) {
  __shared__ float a[64];
  a[threadIdx.x] = (threadIdx.x < 48) ? pen[threadIdx.x] : 0.f;
  __syncthreads();
  for (int st = 32; st > 0; st >>= 1) {
    if ((int)threadIdx.x < st) a[threadIdx.x] += a[threadIdx.x + st];
    __syncthreads();
  }
  if (threadIdx.x == 0)
    out[(size_t)B_ * L_ + (size_t)B_ * L_ * 12] = a[0] / 48.f;  // mean over b, /3 nets
}

// ---------------- host launcher ----------------
extern "C" void kernel_launch(void* const* d_in, const int* in_sizes, int n_in,
                              void* d_out, int out_size, void* d_ws, size_t ws_size,
                              hipStream_t stream) {
  (void)in_sizes; (void)n_in; (void)out_size; (void)ws_size;
  const float* x = (const float*)d_in[0];
  // params flattened in setup_inputs() insertion order:
  // omega[0..1], gamma[0..1], d[0..1] : 10 arrays each layer, then heads, then dt, idx
  auto layer = [&](int net, int lyr) -> LayerP {
    int base = 1 + (net * 2 + lyr) * 10;
    LayerP p;
    p.norm_w     = (const float*)d_in[base + 0];
    p.in_proj_w  = (const float*)d_in[base + 1];
    p.conv_w     = (const float*)d_in[base + 2];
    p.conv_b     = (const float*)d_in[base + 3];
    p.x_proj_w   = (const float*)d_in[base + 4];
    p.dt_proj_w  = (const float*)d_in[base + 5];
    p.dt_proj_b  = (const float*)d_in[base + 6];
    p.A_log      = (const float*)d_in[base + 7];
    p.Dv         = (const float*)d_in[base + 8];
    p.out_proj_w = (const float*)d_in[base + 9];
    return p;
  };
  const float* wo  = (const float*)d_in[61];
  const float* bo  = (const float*)d_in[62];
  const float* wg  = (const float*)d_in[63];
  const float* bg  = (const float*)d_in[64];
  const float* wd  = (const float*)d_in[65];
  const float* bd  = (const float*)d_in[66];
  const float* bsc = (const float*)d_in[67];
  const float* dtp = (const float*)d_in[68];

  float* ws = (float*)d_ws;
  size_t off = 0;
  auto alloc = [&](size_t nfl) { float* p = ws + off; off += nfl; return p; };
  float* z1   = alloc((size_t)B_ * L_);
  float* xdot = alloc((size_t)B_ * L_);
  NetBufs nb;
  for (int net = 0; net < 3; ++net) {
    nb.u[net]    = alloc((size_t)B_ * L_ * US);
    nb.pz[net]   = alloc((size_t)B_ * L_ * PS);
    nb.xi[net]   = alloc((size_t)B_ * L_ * 4);
    nb.sbuf[net] = alloc((size_t)B_ * L_ * SS);
    nb.y[net]    = alloc((size_t)B_ * L_ * 4);
  }
  float* ho  = alloc((size_t)B_ * L_);
  float* hg  = alloc((size_t)B_ * L_);
  float* hd  = alloc((size_t)B_ * L_);
  float* pen = alloc(64);

  k_prep<<<dim3(L_ / 256, B_), 256, 0, stream>>>(x, z1, xdot, nb.u[0], nb.u[1], nb.u[2]);

  for (int lyr = 0; lyr < 2; ++lyr) {
    LayerP3 P;
    for (int net = 0; net < 3; ++net) P.p[net] = layer(net, lyr);
    k_normproj<<<dim3(TILES, B_, 3), 32, 0, stream>>>(P, nb);
    k_convproj<<<dim3(TILES, B_, 3), 32, 0, stream>>>(P, nb);
    k_scan   <<<dim3(B_, 3),        64, 0, stream>>>(P, nb);
    k_outproj<<<dim3(TILES, B_, 3), 32, 0, stream>>>(P, nb);
  }

  k_heads<<<dim3(L_ / 256, B_), 256, 0, stream>>>(nb.u[0], nb.u[1], nb.u[2], z1, xdot,
                                                  wo, bo, wg, bg, wd, bd, bsc, dtp,
                                                  (float*)d_out, ho, hg, hd);
  k_pen1<<<dim3(B_, 3), 256, 0, stream>>>(ho, hg, hd, pen);
  k_pen2<<<1, 64, 0, stream>>>(pen, (float*)d_out);
}